// SetAbstractionLayer_56727928045596
// MI455X (gfx1250) — compile-verified
//
#include <hip/hip_runtime.h>
#include <hip/hip_bf16.h>

typedef __attribute__((ext_vector_type(2))) float v2f;
typedef __attribute__((ext_vector_type(8))) float v8f;

#define NPTS   8192
#define NBATCH 8
#define NSAMP  2048
#define KNBR   32
#define DFEAT  32
#define MTOT   (NBATCH * NSAMP * KNBR)   // 524288

// ---------------------------------------------------------------------------
// 1) Farthest point sampling: one block per batch, 1024 threads, 8 pts/thread
//    in registers. Matches jnp.argmax first-index tie-break.
// ---------------------------------------------------------------------------
__global__ __launch_bounds__(1024) void fps_kernel(const float* __restrict__ xyz,
                                                   float* __restrict__ new_xyz) {
  __shared__ float rv[32];
  __shared__ int   ri[32];
  __shared__ int   cur;
  const int b = blockIdx.x;
  const int tid = threadIdx.x;
  const int lane = tid & 31, wv = tid >> 5;
  const float* base = xyz + (size_t)b * NPTS * 3;

  float px[8], py[8], pz[8], dist[8];
#pragma unroll
  for (int j = 0; j < 8; ++j) {
    int i = j * 1024 + tid;
    px[j] = base[i * 3 + 0];
    py[j] = base[i * 3 + 1];
    pz[j] = base[i * 3 + 2];
    dist[j] = 1e10f;
  }
  if (tid == 0) cur = 0;
  __syncthreads();

  for (int s = 0; s < NSAMP; ++s) {
    const int far = cur;
    const float cx = base[far * 3 + 0];
    const float cy = base[far * 3 + 1];
    const float cz = base[far * 3 + 2];
    if (tid == 0) {
      size_t o = ((size_t)b * NSAMP + s) * 3;
      new_xyz[o + 0] = cx; new_xyz[o + 1] = cy; new_xyz[o + 2] = cz;
    }
    float bvv = -1.0f; int bii = 0x7fffffff;
#pragma unroll
    for (int j = 0; j < 8; ++j) {
      int i = j * 1024 + tid;
      float dx = px[j] - cx, dy = py[j] - cy, dz = pz[j] - cz;
      float d = dx * dx + dy * dy + dz * dz;
      float nd = fminf(dist[j], d);
      dist[j] = nd;
      if (nd > bvv || (nd == bvv && i < bii)) { bvv = nd; bii = i; }
    }
#pragma unroll
    for (int off = 16; off >= 1; off >>= 1) {
      float ov = __shfl_xor(bvv, off, 32);
      int   oi = __shfl_xor(bii, off, 32);
      if (ov > bvv || (ov == bvv && oi < bii)) { bvv = ov; bii = oi; }
    }
    if (lane == 0) { rv[wv] = bvv; ri[wv] = bii; }
    __syncthreads();
    if (wv == 0) {
      float v = rv[lane]; int idx = ri[lane];
#pragma unroll
      for (int off = 16; off >= 1; off >>= 1) {
        float ov = __shfl_xor(v, off, 32);
        int   oi = __shfl_xor(idx, off, 32);
        if (ov > v || (ov == v && oi < idx)) { v = ov; idx = oi; }
      }
      if (lane == 0) cur = idx;
    }
    __syncthreads();
  }
}

// ---------------------------------------------------------------------------
// 2) Ball query + grouping: one wave per query point. Ballot-scan over the
//    8192 points in index order -> first K in-radius indices (== sort path).
//    Writes grouped features channel-major: X[c*M + (q*K + slot)].
// ---------------------------------------------------------------------------
__global__ __launch_bounds__(256) void group_kernel(const float* __restrict__ xyz,
                                                    const float* __restrict__ points,
                                                    const float* __restrict__ new_xyz,
                                                    float* __restrict__ X, int M) {
  const int lane = threadIdx.x & 31;
  const int wv = threadIdx.x >> 5;
  const int q = blockIdx.x * 8 + wv;       // [0, B*S)
  const int b = q >> 11;                   // q / 2048
  const float* bx = xyz + (size_t)b * NPTS * 3;
  const float* bp = points + (size_t)b * NPTS * DFEAT;
  const float qx = new_xyz[q * 3 + 0];
  const float qy = new_xyz[q * 3 + 1];
  const float qz = new_xyz[q * 3 + 2];
  const float r2 = 0.01f;

  int count = 0, firstIdx = -1;
  for (int step = 0; step < NPTS / 32; ++step) {
    int i = step * 32 + lane;
    float dx = bx[i * 3 + 0] - qx;
    float dy = bx[i * 3 + 1] - qy;
    float dz = bx[i * 3 + 2] - qz;
    float d2 = dx * dx + dy * dy + dz * dz;
    bool pred = (d2 <= r2);
    unsigned mask = (unsigned)__ballot((int)pred);
    if (mask) {
      if (firstIdx < 0) firstIdx = step * 32 + (__ffs(mask) - 1);
      int pos = count + __popc(mask & ((1u << lane) - 1u));
      if (pred && pos < KNBR) {
        size_t mcol = (size_t)q * KNBR + pos;
        X[(size_t)0 * M + mcol] = dx;
        X[(size_t)1 * M + mcol] = dy;
        X[(size_t)2 * M + mcol] = dz;
#pragma unroll
        for (int c = 0; c < DFEAT; ++c)
          X[(size_t)(3 + c) * M + mcol] = bp[(size_t)i * DFEAT + c];
      }
      count += __popc(mask);
      if (count >= KNBR) break;
    }
  }
  // pad remaining slots (slot == lane) with the first neighbor's data
  int total = count < KNBR ? count : KNBR;
  size_t mcol = (size_t)q * KNBR + lane;
  if (lane >= total) {
    int i = (firstIdx >= 0) ? firstIdx : (NPTS - 1);
    float dx = bx[i * 3 + 0] - qx;
    float dy = bx[i * 3 + 1] - qy;
    float dz = bx[i * 3 + 2] - qz;
    X[(size_t)0 * M + mcol] = dx;
    X[(size_t)1 * M + mcol] = dy;
    X[(size_t)2 * M + mcol] = dz;
#pragma unroll
    for (int c = 0; c < DFEAT; ++c)
      X[(size_t)(3 + c) * M + mcol] = bp[(size_t)i * DFEAT + c];
  }
  X[(size_t)35 * M + mcol] = 0.0f;   // pad channel for K=4 WMMA stepping
}

// ---------------------------------------------------------------------------
// 3) WMMA fp32 GEMM: Y[o,m] = sum_c W[o,c] * f(X[c,m]) + bias[o]
//    f = identity (layer 0) or ReLU(x*scale[c]+shift[c]) (BN of prev layer).
//    Also accumulates per-channel sum / sumsq for this layer's BN.
// ---------------------------------------------------------------------------
template <int CIN, int COUT, bool BN_IN, int MT>
__global__ __launch_bounds__(256) void gemm_bn_kernel(const float* __restrict__ X,
                                                      const float* __restrict__ W,
                                                      const float* __restrict__ bias,
                                                      const float* __restrict__ inScale,
                                                      const float* __restrict__ inShift,
                                                      float* __restrict__ Y,
                                                      float* __restrict__ sumBuf,
                                                      float* __restrict__ sqBuf,
                                                      int M) {
  constexpr int CK = (CIN + 3) / 4;       // K-steps of 4
  constexpr int OT = COUT / 16;           // output-row tiles
  constexpr int MW = 8 / OT;              // concurrent m-slots per block
  const int wv = threadIdx.x >> 5;
  const int lane = threadIdx.x & 31;
  const int hl = lane >> 4;               // lane half
  const int l = lane & 15;
  const int o_tile = wv % OT;
  const int mslot = wv / OT;
  const int o_base = o_tile * 16;
  const int m_block0 = blockIdx.x * (MW * MT * 16);

  // Preload A fragments (weights) for all K-steps.
  float a0[CK], a1[CK];
#pragma unroll
  for (int ct = 0; ct < CK; ++ct) {
    int c0 = 4 * ct + 2 * hl;
    int row = o_base + l;
    a0[ct] = (c0 < CIN) ? W[row * CIN + c0] : 0.0f;
    a1[ct] = (c0 + 1 < CIN) ? W[row * CIN + c0 + 1] : 0.0f;
  }
  float bv[8];
#pragma unroll
  for (int v = 0; v < 8; ++v) bv[v] = bias[o_base + hl * 8 + v];

  float s1[8], s2[8];
#pragma unroll
  for (int v = 0; v < 8; ++v) { s1[v] = 0.0f; s2[v] = 0.0f; }

  for (int mt = 0; mt < MT; ++mt) {
    const int m0 = m_block0 + (mslot * MT + mt) * 16;
    v8f acc = {0.f, 0.f, 0.f, 0.f, 0.f, 0.f, 0.f, 0.f};
#pragma unroll
    for (int ct = 0; ct < CK; ++ct) {
      int c0 = 4 * ct + 2 * hl;
      float x0 = 0.0f, x1 = 0.0f;
      if (c0 < CIN) {
        x0 = X[(size_t)c0 * M + m0 + l];
        if (BN_IN) { x0 = x0 * inScale[c0] + inShift[c0]; x0 = x0 > 0.0f ? x0 : 0.0f; }
      }
      if (c0 + 1 < CIN) {
        x1 = X[(size_t)(c0 + 1) * M + m0 + l];
        if (BN_IN) { x1 = x1 * inScale[c0 + 1] + inShift[c0 + 1]; x1 = x1 > 0.0f ? x1 : 0.0f; }
      }
      v2f Af = {a0[ct], a1[ct]};
      v2f Bf = {x0, x1};
      acc = __builtin_amdgcn_wmma_f32_16x16x4_f32(false, Af, false, Bf,
                                                  (short)0, acc, false, false);
    }
#pragma unroll
    for (int v = 0; v < 8; ++v) {
      float y = acc[v] + bv[v];
      int o = o_base + hl * 8 + v;
      Y[(size_t)o * M + m0 + l] = y;
      s1[v] += y;
      s2[v] += y * y;
    }
  }
  // Reduce stats across the 16 lanes sharing each channel, then atomics.
#pragma unroll
  for (int off = 8; off >= 1; off >>= 1) {
#pragma unroll
    for (int v = 0; v < 8; ++v) {
      s1[v] += __shfl_xor(s1[v], off, 32);
      s2[v] += __shfl_xor(s2[v], off, 32);
    }
  }
  if (l == 0) {
#pragma unroll
    for (int v = 0; v < 8; ++v) {
      int o = o_base + hl * 8 + v;
      atomicAdd(&sumBuf[o], s1[v]);
      atomicAdd(&sqBuf[o], s2[v]);
    }
  }
}

// ---------------------------------------------------------------------------
// 4) BN scale/shift from accumulated sums (training-mode batch stats).
// ---------------------------------------------------------------------------
__global__ void bn_prep_kernel(const float* __restrict__ sum, const float* __restrict__ sq,
                               const float* __restrict__ g, const float* __restrict__ be,
                               float* __restrict__ scale, float* __restrict__ shift,
                               int C, float invM) {
  int c = threadIdx.x;
  if (c < C) {
    float mu = sum[c] * invM;
    float var = sq[c] * invM - mu * mu;
    float s = g[c] * rsqrtf(var + 1e-5f);
    scale[c] = s;
    shift[c] = be[c] - mu * s;
  }
}

__global__ void zero_stats_kernel(float* p) { p[threadIdx.x] = 0.0f; }

// ---------------------------------------------------------------------------
// 5) Fused BN+ReLU+max-pool over K, transposed output (B,S,128).
// ---------------------------------------------------------------------------
__global__ __launch_bounds__(128) void maxpool_kernel(const float* __restrict__ Y,
                                                      const float* __restrict__ scale,
                                                      const float* __restrict__ shift,
                                                      float* __restrict__ out, int M) {
  const int q = blockIdx.x;     // b*S + s
  const int o = threadIdx.x;    // channel 0..127
  const float sc = scale[o], sh = shift[o];
  const float* row = Y + (size_t)o * M + (size_t)q * KNBR;
  float best = 0.0f;            // ReLU outputs are >= 0
#pragma unroll
  for (int k = 0; k < KNBR; ++k) {
    float v = row[k] * sc + sh;
    v = v > 0.0f ? v : 0.0f;
    best = fmaxf(best, v);
  }
  out[(size_t)NBATCH * NSAMP * 3 + (size_t)q * 128 + o] = best;
}

// ---------------------------------------------------------------------------
extern "C" void kernel_launch(void* const* d_in, const int* in_sizes, int n_in,
                              void* d_out, int out_size, void* d_ws, size_t ws_size,
                              hipStream_t stream) {
  const float* xyz = (const float*)d_in[0];
  const float* points = (const float*)d_in[1];
  const float* W0 = (const float*)d_in[2];
  const float* b0 = (const float*)d_in[3];
  const float* g0 = (const float*)d_in[4];
  const float* be0 = (const float*)d_in[5];
  const float* W1 = (const float*)d_in[6];
  const float* b1 = (const float*)d_in[7];
  const float* g1 = (const float*)d_in[8];
  const float* be1 = (const float*)d_in[9];
  const float* W2 = (const float*)d_in[10];
  const float* b2 = (const float*)d_in[11];
  const float* g2 = (const float*)d_in[12];
  const float* be2 = (const float*)d_in[13];
  float* out = (float*)d_out;

  const int M = MTOT;
  float* bufA = (float*)d_ws;                   // 64*M floats: X0 (36*M) then Y1
  float* bufB = bufA + (size_t)64 * M;          // 64*M floats: Y0
  float* bufC = bufB + (size_t)64 * M;          // 128*M floats: Y2
  float* stats = bufC + (size_t)128 * M;
  float* sum0 = stats +   0; float* sq0 = stats +  64;
  float* sum1 = stats + 128; float* sq1 = stats + 192;
  float* sum2 = stats + 256; float* sq2 = stats + 384;  // 128 ch
  float* sc0 = stats + 512; float* sh0 = stats + 576;
  float* sc1 = stats + 640; float* sh1 = stats + 704;
  float* sc2 = stats + 768; float* sh2 = stats + 896;   // 128 ch

  zero_stats_kernel<<<1, 512, 0, stream>>>(stats);

  fps_kernel<<<NBATCH, 1024, 0, stream>>>(xyz, out);

  group_kernel<<<NBATCH * NSAMP / 8, 256, 0, stream>>>(xyz, points, out, bufA, M);

  gemm_bn_kernel<35, 64, false, 8><<<M / 256, 256, 0, stream>>>(
      bufA, W0, b0, nullptr, nullptr, bufB, sum0, sq0, M);
  bn_prep_kernel<<<1, 64, 0, stream>>>(sum0, sq0, g0, be0, sc0, sh0, 64, 1.0f / M);

  gemm_bn_kernel<64, 64, true, 8><<<M / 256, 256, 0, stream>>>(
      bufB, W1, b1, sc0, sh0, bufA, sum1, sq1, M);
  bn_prep_kernel<<<1, 64, 0, stream>>>(sum1, sq1, g1, be1, sc1, sh1, 64, 1.0f / M);

  gemm_bn_kernel<64, 128, true, 8><<<M / 128, 256, 0, stream>>>(
      bufA, W2, b2, sc1, sh1, bufC, sum2, sq2, M);
  bn_prep_kernel<<<1, 128, 0, stream>>>(sum2, sq2, g2, be2, sc2, sh2, 128, 1.0f / M);

  maxpool_kernel<<<NBATCH * NSAMP, 128, 0, stream>>>(bufC, sc2, sh2, out, M);
}